// CayleyConv_52243982189166
// MI455X (gfx1250) — compile-verified
//
#include <hip/hip_runtime.h>
#include <hip/hip_bf16.h>

#define NN   4096
#define NE   200000     // directed edges after concat (2*100000)
#define CIN  64
#define COUT 64

typedef float v2f __attribute__((ext_vector_type(2)));
typedef float v4f __attribute__((ext_vector_type(4)));
typedef float v8f __attribute__((ext_vector_type(8)));

// ---------------- init: zero L (used as A accumulator), deg, Ublk, state; mark = -1
__global__ void k_init(float* L, float* deg, int* mark, float* Ublk, int* state) {
    long i = (long)blockIdx.x * blockDim.x + threadIdx.x;
    if (i < (long)NN * NN / 4) ((v4f*)L)[i] = (v4f){0.f, 0.f, 0.f, 0.f};
    if (i < NN) { deg[i] = 0.0f; mark[i] = -1; }
    if (i < 32 * 32) Ublk[i] = 0.0f;
    if (i < 64) state[i] = 0;
}

// ---------------- scatter edges: A[r,c] += 1, deg[r] += 1 (exact float int adds)
__global__ void k_scatter(const long long* ei, float* L, float* deg) {
    int e = blockIdx.x * blockDim.x + threadIdx.x;
    if (e >= NE) return;
    int r = (int)ei[e];
    int c = (int)ei[NE + e];
    atomicAdd(&L[(long)r * NN + c], 1.0f);
    atomicAdd(&deg[r], 1.0f);
}

__global__ void k_dinv(const float* deg, float* dinv) {
    int i = blockIdx.x * blockDim.x + threadIdx.x;
    if (i >= NN) return;
    float d = deg[i];
    dinv[i] = (d > 0.0f) ? (1.0f / sqrtf(d)) : 0.0f;
}

// ---------------- L = I - dinv_i * A * dinv_j  (in place, B128 traffic)
__global__ void k_lap(float* L, const float* dinv) {
    long idx4 = (long)blockIdx.x * blockDim.x + threadIdx.x;
    if (idx4 >= (long)NN * NN / 4) return;
    int i  = (int)(idx4 >> 10);               // row (4096/4 vectors per row)
    int j0 = (int)((idx4 << 2) & (NN - 1));   // first column of the vector
    v4f a  = ((v4f*)L)[idx4];
    float di = dinv[i];
    v4f dj = ((const v4f*)dinv)[j0 >> 2];
    v4f r;
#pragma unroll
    for (int e = 0; e < 4; e++)
        r[e] = ((i == j0 + e) ? 1.0f : 0.0f) - di * a[e] * dj[e];
    ((v4f*)L)[idx4] = r;
}

// ---------------- argmax over upper triangle of |L|, first-index tie-break, B128 loads
__global__ void k_amax1(const float* L, float* pV, int* pI) {
    __shared__ float sv[256];
    __shared__ int   si[256];
    int tid = threadIdx.x;
    long base = (long)blockIdx.x * 4096;
    const v4f* L4 = (const v4f*)(L + base);
    float best = -1.0f; int bidx = 0x7fffffff;
    for (int t = tid; t < 1024; t += 256) {
        v4f v4 = L4[t];
        int idx0 = (int)base + t * 4;
        int i  = idx0 >> 12;                  // rows are 4096 wide; vector never crosses
        int j0 = idx0 & (NN - 1);
#pragma unroll
        for (int e = 0; e < 4; e++) {
            float v = (j0 + e > i) ? fabsf(v4[e]) : 0.0f;  // == jnp.where(upper,|a|,0)
            int ii = idx0 + e;
            if (v > best || (v == best && ii < bidx)) { best = v; bidx = ii; }
        }
    }
    sv[tid] = best; si[tid] = bidx;
    __syncthreads();
    for (int s = 128; s > 0; s >>= 1) {
        if (tid < s) {
            float v2 = sv[tid + s]; int i2 = si[tid + s];
            if (v2 > sv[tid] || (v2 == sv[tid] && i2 < si[tid])) { sv[tid] = v2; si[tid] = i2; }
        }
        __syncthreads();
    }
    if (tid == 0) { pV[blockIdx.x] = sv[0]; pI[blockIdx.x] = si[0]; }
}

// stage2: reduce partials -> (k,l); compute (c,s) per reference; update compact U block
__global__ void k_amax2(const float* pV, const int* pI, const float* L,
                        int* state, float* Ublk, int* mark) {
    __shared__ float sv[256];
    __shared__ int   si[256];
    int tid = threadIdx.x;
    float best = -1.0f; int bidx = 0x7fffffff;
    for (int b = tid; b < 4096; b += 256) {
        float v = pV[b]; int i = pI[b];
        if (v > best || (v == best && i < bidx)) { best = v; bidx = i; }
    }
    sv[tid] = best; si[tid] = bidx;
    __syncthreads();
    for (int s = 128; s > 0; s >>= 1) {
        if (tid < s) {
            float v2 = sv[tid + s]; int i2 = si[tid + s];
            if (v2 > sv[tid] || (v2 == sv[tid] && i2 < si[tid])) { sv[tid] = v2; si[tid] = i2; }
        }
        __syncthreads();
    }
    if (tid == 0) {
        int idx = si[0];
        int k = idx >> 12;
        int l = idx & (NN - 1);
        float akk = L[(long)k * NN + k];
        float all = L[(long)l * NN + l];
        float akl = L[(long)k * NN + l];
        float aDiff = all - akk;
        float akl_safe   = (akl   == 0.0f) ? 1.0f : akl;
        float aDiff_safe = (aDiff == 0.0f) ? 1.0f : aDiff;
        float phi = aDiff / (2.0f * akl_safe);
        float t2 = 1.0f / (fabsf(phi) + sqrtf(phi * phi + 1.0f));
        if (phi < 0.0f) t2 = -t2;
        float t1 = akl / aDiff_safe;
        float t = (fabsf(akl) < fabsf(aDiff) * 1e-36f) ? t1 : t2;
        float cc = 1.0f / sqrtf(t * t + 1.0f);
        float ss = t * cc;
        state[0] = k; state[1] = l;
        ((float*)state)[2] = cc;
        ((float*)state)[3] = ss;
        // --- compact U (identity + dense block on touched indices) ---
        int tc = state[4];
        int* tlist = state + 8;
        int ik = mark[k];
        if (ik < 0) { ik = tc; tlist[tc] = k; mark[k] = tc; Ublk[tc * 32 + tc] = 1.0f; tc++; }
        int il = mark[l];
        if (il < 0) { il = tc; tlist[tc] = l; mark[l] = tc; Ublk[tc * 32 + tc] = 1.0f; tc++; }
        state[4] = tc;
        for (int a = 0; a < tc; a++) {
            float pk = Ublk[a * 32 + ik], pl = Ublk[a * 32 + il];
            Ublk[a * 32 + ik] = cc * pk - ss * pl;
            Ublk[a * 32 + il] = ss * pk + cc * pl;
        }
    }
}

// ---------------- Givens rotation: columns then rows (separate launches for ordering)
__global__ void k_rotc(float* L, const int* state) {
    int i = blockIdx.x * blockDim.x + threadIdx.x;
    if (i >= NN) return;
    int k = state[0], l = state[1];
    float c = ((const float*)state)[2], s = ((const float*)state)[3];
    float ck = L[(long)i * NN + k], cl = L[(long)i * NN + l];
    L[(long)i * NN + k] = c * ck - s * cl;
    L[(long)i * NN + l] = s * ck + c * cl;
}
__global__ void k_rotr(float* L, const int* state) {
    int j = blockIdx.x * blockDim.x + threadIdx.x;
    if (j >= NN) return;
    int k = state[0], l = state[1];
    float c = ((const float*)state)[2], s = ((const float*)state)[3];
    float rk = L[(long)k * NN + j], rl = L[(long)l * NN + j];
    L[(long)k * NN + j] = (j == l) ? 0.0f : (c * rk - s * rl);
    L[(long)l * NN + j] = (j == k) ? 0.0f : (s * rk + c * rl);
}

// ---------------- U-block GEMM via V_WMMA_F32_16X16X4_F32 (branch-free, templated):
// dst(32x64) = A(32x32) * B(32x64), A = W (TRANSPOSE=0) or W^T (TRANSPOSE=1);
// B gathered via tlist (zero-padded). 8 waves, one 16x16 tile each; 8 WMMAs per wave.
template <int TRANSPOSE>
__global__ void k_ugemm(const float* W, const float* src, const int* state,
                        float* dst, long rs, long cs) {
    __shared__ float sW[32 * 32];
    __shared__ float sX[32 * 64];
    int tid = threadIdx.x;
    int tc = state[4];
    const int* tlist = state + 8;
    for (int idx = tid; idx < 1024; idx += 256) sW[idx] = W[idx];
    for (int idx = tid; idx < 2048; idx += 256) {
        int k = idx >> 6, nn = idx & 63;
        sX[idx] = (k < tc) ? src[(long)tlist[k] * rs + (long)nn * cs] : 0.0f;
    }
    __syncthreads();
    int wave = tid >> 5, lane = tid & 31;
    int m0 = (wave >> 2) * 16, n0 = (wave & 3) * 16;
    int mrow = m0 + (lane & 15);
    int half = (lane >> 4) * 2;          // A layout: lanes16-31 hold K+2/K+3
    int ncol = n0 + (lane & 15);
    v8f acc = {0.f, 0.f, 0.f, 0.f, 0.f, 0.f, 0.f, 0.f};
#pragma unroll
    for (int kk = 0; kk < 32; kk += 4) {
        int k0 = kk + half;
        v2f a, b;
        if (TRANSPOSE) { a.x = sW[k0 * 32 + mrow]; a.y = sW[(k0 + 1) * 32 + mrow]; }
        else           { a.x = sW[mrow * 32 + k0]; a.y = sW[mrow * 32 + k0 + 1]; }
        b.x = sX[k0 * 64 + ncol];
        b.y = sX[(k0 + 1) * 64 + ncol];
        acc = __builtin_amdgcn_wmma_f32_16x16x4_f32(false, a, false, b, (short)0, acc,
                                                    false, false);
    }
    int mbase = m0 + ((lane >> 4) << 3); // D layout: VGPR v -> M = v (+8 for hi lanes)
#pragma unroll
    for (int v = 0; v < 8; v++) dst[(mbase + v) * 64 + ncol] = acc[v];
}

// ---------------- spectral filter: out[o,n] = sum_i g(o,i,n)*aux[n,i]
// cos(2*atan(1/x)) = (x^2-1)/(x^2+1); sin = 2x/(x^2+1); angle-addition recurrence over r.
__global__ void k_filter(const float* L, const float* x, const float* auxT,
                         const int* mark, const float* h, const float* rw,
                         const float* iw, const float* cmat, float* out) {
    __shared__ float saux[4][64];
    __shared__ float sw[4];
    int sub = threadIdx.x >> 6;
    int o   = threadIdx.x & 63;
    int node = blockIdx.x * 4 + sub;
    int m = mark[node];
    const float* arow = (m >= 0) ? &auxT[m * 64] : &x[(long)node * 64];
    saux[sub][o] = arow[o];
    if (o == 0) sw[sub] = L[(long)node * NN + node];   // eigenvalue = diag(a)
    __syncthreads();
    float w = sw[sub];
    float acc = 0.0f;
    for (int i = 0; i < 64; i++) {
        float hw = h[o * 64 + i] * w;
        float c1, s1;
        if (hw > 1e-5f) {
            float d = 1.0f / (hw * hw + 1.0f);
            c1 = (hw * hw - 1.0f) * d;
            s1 = 2.0f * hw * d;
        } else {
            c1 = -0.41614684f;   // cos(2.0)
            s1 = 0.90929741f;    // sin(2.0)
        }
        float cr = c1, sr = s1, g = 0.0f;
        const v4f rw4 = ((const v4f*)rw)[o * 64 + i];
        const v4f iw4 = ((const v4f*)iw)[o * 64 + i];
#pragma unroll
        for (int r = 0; r < 4; r++) {
            g += rw4[r] * cr - iw4[r] * sr;
            float cn = cr * c1 - sr * s1;
            sr = sr * c1 + cr * s1;
            cr = cn;
        }
        g += cmat[o * 64 + i];
        acc += g * saux[sub][i];
    }
    out[(long)o * NN + node] = acc;
}

// ---------------- output[i,o] = (U @ out.T)[i,o] + bias[o], sparse-U passthrough
__global__ void k_final(const float* out, const float* ovr, const int* mark,
                        const float* bias, float* y) {
    int idx = blockIdx.x * blockDim.x + threadIdx.x;
    if (idx >= NN * COUT) return;
    int i = idx >> 6, o = idx & 63;
    int m = mark[i];
    float v = (m >= 0) ? ovr[m * 64 + o] : out[(long)o * NN + i];
    y[idx] = v + bias[o];
}

extern "C" void kernel_launch(void* const* d_in, const int* in_sizes, int n_in,
                              void* d_out, int out_size, void* d_ws, size_t ws_size,
                              hipStream_t stream) {
    const float*     x    = (const float*)d_in[0];
    const long long* ei   = (const long long*)d_in[1];   // int64 edge_index (2,200000)
    const float*     rw   = (const float*)d_in[2];
    const float*     iw   = (const float*)d_in[3];
    const float*     h    = (const float*)d_in[4];
    const float*     cmat = (const float*)d_in[5];
    const float*     bias = (const float*)d_in[6];
    float*           y    = (float*)d_out;

    char* ws = (char*)d_ws;
    float* L    = (float*)(ws + 0);          // 64 MB dense Laplacian / working matrix
    float* deg  = (float*)(ws + 67108864);
    float* dinv = (float*)(ws + 67125248);
    int*   mark = (int*)  (ws + 67141632);
    float* pV   = (float*)(ws + 67158016);
    int*   pI   = (int*)  (ws + 67174400);
    int*   state= (int*)  (ws + 67190784);   // k,l,c,s,tcount,tlist[32]
    float* Ublk = (float*)(ws + 67191040);   // 32x32 compact eigenvector block
    float* auxT = (float*)(ws + 67195136);   // 32x64
    float* ovr  = (float*)(ws + 67203328);   // 32x64
    float* outb = (float*)(ws + 67211520);   // 64x4096

    k_init   <<<16384, 256, 0, stream>>>(L, deg, mark, Ublk, state);
    k_scatter<<<(NE + 255) / 256, 256, 0, stream>>>(ei, L, deg);
    k_dinv   <<<16, 256, 0, stream>>>(deg, dinv);
    k_lap    <<<16384, 256, 0, stream>>>(L, dinv);

    for (int it = 0; it < 10; it++) {
        k_amax1<<<4096, 256, 0, stream>>>(L, pV, pI);
        k_amax2<<<1, 256, 0, stream>>>(pV, pI, L, state, Ublk, mark);
        k_rotc <<<16, 256, 0, stream>>>(L, state);
        k_rotr <<<16, 256, 0, stream>>>(L, state);
    }

    // aux override rows: auxT = Ublk^T * X[tlist]   (x row-major: rs=64, cs=1)
    k_ugemm<1><<<1, 256, 0, stream>>>(Ublk, x, state, auxT, 64L, 1L);
    k_filter<<<1024, 256, 0, stream>>>(L, x, auxT, mark, h, rw, iw, cmat, outb);
    // output override rows: ovr = Ublk * out.T[tlist]  (out: rs=1, cs=4096)
    k_ugemm<0><<<1, 256, 0, stream>>>(Ublk, outb, state, ovr, 1L, 4096L);
    k_final<<<1024, 256, 0, stream>>>(outb, ovr, mark, bias, y);
}